// NeuralProgrammer_86466281603720
// MI455X (gfx1250) — compile-verified
//
#include <hip/hip_runtime.h>
#include <hip/hip_bf16.h>

#define H      256
#define QLEN   40
#define NNUM   5
#define NOPS   9
#define COLS   64
#define TSTEPS 4
#define NROWS  200000

typedef float v2f __attribute__((ext_vector_type(2)));
typedef float v8f __attribute__((ext_vector_type(8)));

// ws (float) layout:
//  [0..3]    Sdot[t]  = sum_r rs1_r * colval_r   (atomic accumulated)
//  [4..7]    Scnt[t]  = sum_r rs1_r
//  [8..43]   a_op[t][9]
//  [48..303] a_col[t][64]
//  [304]     l_pivot
//  [305]     g_pivot

// ---------------------------------------------------------------------------
// Kernel A: question RNN + pivots + 4-step selector recurrence (1 block).
// 1024 threads: i = tid&255 is the output element, p = tid>>8 is a 4-way
// K-split partial (reduced through LDS) to cut the serial RNN latency.
// ---------------------------------------------------------------------------
__global__ __launch_bounds__(1024) void np_control(
    const int* __restrict__ iq, const float* __restrict__ qnum,
    const int* __restrict__ lwi, const float* __restrict__ E,
    const float* __restrict__ Wx, const float* __restrict__ Wh,
    const float* __restrict__ bvec, const float* __restrict__ W_op,
    const float* __restrict__ op_emb, const float* __restrict__ W_col,
    const float* __restrict__ col_emb, const float* __restrict__ U,
    const float* __restrict__ W_hc, const float* __restrict__ W_hh,
    float* __restrict__ ws)
{
    __shared__ float sx[H], sh[H], sq[H], shh[H], st[H];
    __shared__ float sqh[2 * H];
    __shared__ float red[4 * H];
    __shared__ float sZ[NNUM][H];
    __shared__ float s_aop[NOPS + 1], s_acol[COLS];

    const int tid = threadIdx.x;
    const int i = tid & (H - 1);
    const int p = tid >> 8;

    if (tid < 8) ws[tid] = 0.0f;          // zero scalar accumulators
    if (p == 0) sh[i] = 0.0f;
    __syncthreads();

    // ---- question RNN: h = tanh(x@Wx + h@Wh + b), 40 steps ----
    for (int s = 0; s < QLEN; ++s) {
        if (p == 0) sx[i] = E[(size_t)iq[s] * H + i];
        __syncthreads();
        float acc = 0.0f;
        const int j0 = 64 * p;
        for (int j = j0; j < j0 + 64; ++j)
            acc += sx[j] * Wx[j * H + i] + sh[j] * Wh[j * H + i];
        red[p * H + i] = acc;
        __syncthreads();
        if (p == 0) {
            const float hv = tanhf(red[i] + red[H + i] + red[2 * H + i] +
                                   red[3 * H + i] + bvec[i]);
            sh[i] = hv;
            for (int n = 0; n < NNUM; ++n)
                if (lwi[n] == s) sZ[n][i] = hv;
        }
        __syncthreads();
    }
    if (p == 0) sq[i] = sh[i];            // q = final state
    __syncthreads();

    // ---- pivots: softmax(Z @ U[0/1]) @ question_numbers ----
    if (tid < NNUM) {
        float zl = 0.0f, zg = 0.0f;
        for (int j = 0; j < H; ++j) {
            zl += sZ[tid][j] * U[j];
            zg += sZ[tid][j] * U[H + j];
        }
        s_aop[tid] = zl;
        s_acol[tid] = zg;
    }
    __syncthreads();
    if (tid == 0) {
        float ml = -1e30f, mg = -1e30f;
        for (int n = 0; n < NNUM; ++n) { ml = fmaxf(ml, s_aop[n]); mg = fmaxf(mg, s_acol[n]); }
        float dl = 0.0f, dg = 0.0f, el[NNUM], eg[NNUM];
        for (int n = 0; n < NNUM; ++n) {
            el[n] = expf(s_aop[n] - ml);  dl += el[n];
            eg[n] = expf(s_acol[n] - mg); dg += eg[n];
        }
        float pl = 0.0f, pg = 0.0f;
        for (int n = 0; n < NNUM; ++n) {
            pl += (el[n] / dl) * qnum[n];
            pg += (eg[n] / dg) * qnum[n];
        }
        ws[304] = pl;  // l_pivot (U[0]=lesser)
        ws[305] = pg;  // g_pivot
    }
    if (p == 0) shh[i] = 0.0f;            // h_hist = 0
    __syncthreads();

    // ---- 4 selector steps (only a_op/a_col/h_hist needed; no table) ----
    for (int t = 0; t < TSTEPS; ++t) {
        if (tid < H) sqh[tid] = sq[tid];
        else if (tid < 2 * H) sqh[tid] = shh[tid - H];
        __syncthreads();

        // tanh(W_op @ qh)
        {
            float a = 0.0f;
            const int j0 = 128 * p;
            for (int j = j0; j < j0 + 128; ++j) a += W_op[i * 2 * H + j] * sqh[j];
            red[p * H + i] = a;
        }
        __syncthreads();
        if (p == 0) st[i] = tanhf(red[i] + red[H + i] + red[2 * H + i] + red[3 * H + i]);
        __syncthreads();
        if (tid < NOPS) {
            float s = 0.0f;
            for (int k = 0; k < H; ++k) s += op_emb[tid * H + k] * st[k];
            s_aop[tid] = s;
        }
        __syncthreads();
        if (tid == 0) {
            float m = -1e30f;
            for (int n = 0; n < NOPS; ++n) m = fmaxf(m, s_aop[n]);
            float d = 0.0f;
            for (int n = 0; n < NOPS; ++n) { s_aop[n] = expf(s_aop[n] - m); d += s_aop[n]; }
            for (int n = 0; n < NOPS; ++n) s_aop[n] /= d;
        }
        __syncthreads();

        // tanh(W_col @ qh)
        {
            float a = 0.0f;
            const int j0 = 128 * p;
            for (int j = j0; j < j0 + 128; ++j) a += W_col[i * 2 * H + j] * sqh[j];
            red[p * H + i] = a;
        }
        __syncthreads();
        if (p == 0) st[i] = tanhf(red[i] + red[H + i] + red[2 * H + i] + red[3 * H + i]);
        __syncthreads();
        if (tid < COLS) {
            float s = 0.0f;
            for (int k = 0; k < H; ++k) s += col_emb[tid * H + k] * st[k];
            s_acol[tid] = s;
        }
        __syncthreads();
        if (tid == 0) {
            float m = -1e30f;
            for (int n = 0; n < COLS; ++n) m = fmaxf(m, s_acol[n]);
            float d = 0.0f;
            for (int n = 0; n < COLS; ++n) { s_acol[n] = expf(s_acol[n] - m); d += s_acol[n]; }
            for (int n = 0; n < COLS; ++n) s_acol[n] /= d;
        }
        __syncthreads();

        if (tid < NOPS) ws[8 + t * NOPS + tid] = s_aop[tid];
        if (tid < COLS) ws[48 + t * COLS + tid] = s_acol[tid];

        // c = [a_op @ op_emb, a_col @ col_emb]
        if (tid < H) {
            float c1 = 0.0f;
            for (int n = 0; n < NOPS; ++n) c1 += s_aop[n] * op_emb[n * H + tid];
            sqh[tid] = c1;
        } else if (tid < 2 * H) {
            float c2 = 0.0f;
            const int ii = tid - H;
            for (int n = 0; n < COLS; ++n) c2 += s_acol[n] * col_emb[n * H + ii];
            sqh[tid] = c2;
        }
        __syncthreads();

        // h_hist = tanh(W_hc @ c + W_hh @ h_hist)
        {
            float a = 0.0f;
            const int j0 = 128 * p;
            for (int j = j0; j < j0 + 128; ++j) a += W_hc[i * 2 * H + j] * sqh[j];
            const int j1 = 64 * p;
            for (int j = j1; j < j1 + 64; ++j) a += W_hh[i * H + j] * shh[j];
            red[p * H + i] = a;
        }
        __syncthreads();
        if (p == 0) shh[i] = tanhf(red[i] + red[H + i] + red[2 * H + i] + red[3 * H + i]);
        __syncthreads();
    }
}

// ---------------------------------------------------------------------------
// Kernel B: single fused pass over the table. Per wave: tiles of 16 rows.
// D[t][row] = sum_c A[t][c]*B[c][row] via V_WMMA_F32_16X16X4_F32 (16 K-chunks,
// 3 shared-A accumulators: raw / >g_pivot / <l_pivot). D lands as dot[t] in
// VGPR t, lane=row -> the row_select recurrence is per-lane math in lanes
// 0..15. Loads are batched per tile (one clause, one wait) with a prefetch of
// the wave's next tile; the lookup write is a fully coalesced 32-lane stream.
// ---------------------------------------------------------------------------
__global__ __launch_bounds__(256) void np_table(
    const float* __restrict__ table, float* __restrict__ ws,
    float* __restrict__ out)
{
    __shared__ float s_acol[TSTEPS * COLS];
    __shared__ float s_aop[TSTEPS * NOPS];

    const int tid = threadIdx.x;
    if (tid < TSTEPS * COLS) s_acol[tid] = ws[48 + tid];
    if (tid < TSTEPS * NOPS) s_aop[tid] = ws[8 + tid];
    __syncthreads();

    const float lp = ws[304];
    const float gp = ws[305];

    const int lane = tid & 31;
    const int m = lane & 15;           // A row (=t) / B col (=row-in-tile)
    const int hi = lane >> 4;          // K half: offsets {0,1} or {2,3}
    const int wid = (tid >> 5) + blockIdx.x * (blockDim.x >> 5);
    const int nw = (gridDim.x * blockDim.x) >> 5;
    const float lanemask = (lane < 16) ? 1.0f : 0.0f;
    const float aop37 = s_aop[3 * NOPS + 7];
    const float ac3a = s_acol[3 * COLS + lane];        // a_col[3][lane]
    const float ac3b = s_acol[3 * COLS + 32 + lane];   // a_col[3][32+lane]

    // A operand per K-chunk: row M=t (t<4) holds a_col[t][chunk K-slice]
    v2f Achunk[16];
#pragma unroll
    for (int kc = 0; kc < 16; ++kc) {
        float a0 = 0.0f, a1 = 0.0f;
        if (m < TSTEPS) {
            a0 = s_acol[m * COLS + kc * 4 + 2 * hi + 0];
            a1 = s_acol[m * COLS + kc * 4 + 2 * hi + 1];
        }
        Achunk[kc].x = a0;
        Achunk[kc].y = a1;
    }

    float sdot[TSTEPS] = {0.f, 0.f, 0.f, 0.f};
    float scnt[TSTEPS] = {0.f, 0.f, 0.f, 0.f};
    const int ntiles = NROWS / 16;     // 12500, exact

    for (int tile = wid; tile < ntiles; tile += nw) {
        const int row = tile * 16 + m;
        const float* trow = table + (size_t)row * COLS + 2 * hi;

        // prefetch this lane's slice of the wave's next tile
        {
            const int pt = (tile + nw < ntiles) ? (tile + nw) : tile;
            __builtin_prefetch(table + (size_t)(pt * 16 + m) * COLS + 2 * hi, 0, 1);
        }

        // ---- load phase: whole 16-row x 64-col tile slice, batched ----
        v2f bvv[16];
#pragma unroll
        for (int kc = 0; kc < 16; ++kc)
            bvv[kc] = __builtin_nontemporal_load(
                reinterpret_cast<const v2f*>(trow + kc * 4));

        // ---- compute phase: masks + 48 WMMAs ----
        v8f accV = {0.f, 0.f, 0.f, 0.f, 0.f, 0.f, 0.f, 0.f};
        v8f accG = {0.f, 0.f, 0.f, 0.f, 0.f, 0.f, 0.f, 0.f};
        v8f accL = {0.f, 0.f, 0.f, 0.f, 0.f, 0.f, 0.f, 0.f};
#pragma unroll
        for (int kc = 0; kc < 16; ++kc) {
            const v2f bv = bvv[kc];
            v2f Bg; Bg.x = (bv.x > gp) ? 1.f : 0.f;    Bg.y = (bv.y > gp) ? 1.f : 0.f;
            v2f Bl; Bl.x = (bv.x < lp) ? 1.f : 0.f;    Bl.y = (bv.y < lp) ? 1.f : 0.f;
            accV = __builtin_amdgcn_wmma_f32_16x16x4_f32(false, Achunk[kc], false, bv,
                                                         (short)0, accV, false, false);
            accG = __builtin_amdgcn_wmma_f32_16x16x4_f32(false, Achunk[kc], false, Bg,
                                                         (short)0, accG, false, false);
            accL = __builtin_amdgcn_wmma_f32_16x16x4_f32(false, Achunk[kc], false, Bl,
                                                         (short)0, accL, false, false);
        }

        // lanes 0..15: accX[t] == product for time step t, row = tile*16+lane
        float rs1 = 1.0f, rs2 = 1.0f;
#pragma unroll
        for (int t = 0; t < TSTEPS; ++t) {
            const float dv = accV[t], dg = accG[t], dl = accL[t];
            const float* ao = &s_aop[t * NOPS];
            const float rsel = ao[3] * dg + ao[4] * dl
                             + ao[5] * fminf(rs1, rs2) + ao[6] * fmaxf(rs1, rs2)
                             + ao[8] + (ao[0] + ao[1] + ao[2] + ao[7]) * rs1;
            sdot[t] += lanemask * rs1 * dv;
            scnt[t] += lanemask * rs1;
            rs2 = rs1;
            rs1 = rsel;
        }

        // ---- lookup write, fully coalesced across all 32 lanes ----
        // lookup[row][col] = aop3[7] * row_select3[row] * a_col3[col]
        // element e = k*32+lane covers row k/2, col (k&1)*32+lane.
        const float wsel = aop37 * rs1;   // valid in lanes 0..15 (row = lane)
        float* obase = out + 1 + (size_t)tile * (16 * COLS) + lane;
#pragma unroll
        for (int k = 0; k < 32; ++k) {
            const float wk = __shfl(wsel, k >> 1, 32);
            const float v = wk * ((k & 1) ? ac3b : ac3a);
            __builtin_nontemporal_store(v, obase + k * 32);
        }
    }

    // reduce lanes 0..15 (lanes 16..31 masked to zero) and accumulate
#pragma unroll
    for (int t = 0; t < TSTEPS; ++t) {
        float a = sdot[t], c = scnt[t];
        for (int off = 8; off > 0; off >>= 1) {
            a += __shfl_xor(a, off, 32);
            c += __shfl_xor(c, off, 32);
        }
        if (lane == 0) {
            atomicAdd(&ws[t], a);
            atomicAdd(&ws[4 + t], c);
        }
    }
}

// ---------------------------------------------------------------------------
// Kernel C: sequential 4-step scalar chain (needs global sums), writes out[0]
// ---------------------------------------------------------------------------
__global__ void np_finalize(const float* __restrict__ ws, float* __restrict__ out)
{
    if (threadIdx.x == 0 && blockIdx.x == 0) {
        float scal[TSTEPS + 1];
        scal[0] = 0.0f;
        for (int t = 0; t < TSTEPS; ++t) {
            const float s1 = scal[t];
            const float s3 = scal[(t >= 2) ? (t - 2) : 0];
            const float* ao = &ws[8 + t * NOPS];
            scal[t + 1] = ao[0] * ws[t] + ao[1] * ws[4 + t] + ao[2] * (s3 - s1);
        }
        out[0] = scal[TSTEPS];
    }
}

extern "C" void kernel_launch(void* const* d_in, const int* in_sizes, int n_in,
                              void* d_out, int out_size, void* d_ws, size_t ws_size,
                              hipStream_t stream) {
    const int*   iq      = (const int*)d_in[0];
    const float* qnum    = (const float*)d_in[1];
    const int*   lwi     = (const int*)d_in[2];
    const float* table   = (const float*)d_in[3];
    const float* E       = (const float*)d_in[4];
    const float* Wx      = (const float*)d_in[5];
    const float* Wh      = (const float*)d_in[6];
    const float* bvec    = (const float*)d_in[7];
    const float* W_op    = (const float*)d_in[8];
    const float* op_emb  = (const float*)d_in[9];
    const float* W_col   = (const float*)d_in[10];
    const float* col_emb = (const float*)d_in[11];
    const float* U       = (const float*)d_in[12];
    const float* W_hc    = (const float*)d_in[13];
    const float* W_hh    = (const float*)d_in[14];
    float* out = (float*)d_out;
    float* ws  = (float*)d_ws;

    np_control<<<1, 1024, 0, stream>>>(iq, qnum, lwi, E, Wx, Wh, bvec, W_op,
                                       op_emb, W_col, col_emb, U, W_hc, W_hh, ws);
    np_table<<<512, 256, 0, stream>>>(table, ws, out);
    np_finalize<<<1, 1, 0, stream>>>(ws, out);
}